// InvariantPointAttentionLayer_5437428597144
// MI455X (gfx1250) — compile-verified
//
#include <hip/hip_runtime.h>
#include <math.h>

// ---------------------------------------------------------------------------
// Invariant Point Attention for MI455X (gfx1250), fp32 WMMA 16x16x4 pipeline.
// Memory-bound on `e` (268 MB, streamed twice); all GEMMs via v_wmma_f32_16x16x4_f32.
// ---------------------------------------------------------------------------

typedef __attribute__((ext_vector_type(2))) float v2f;
typedef __attribute__((ext_vector_type(8))) float v8f;

#define NH    8
#define LQ    512
#define DRES  384
#define DPAIR 128

constexpr float SS = 0.25f;                 // D_SPH^-0.5 = 16^-0.5
constexpr float SP = 0.23570226039551584f;  // (4.5*4)^-0.5 = 1/sqrt(18)
constexpr float ST = 0.5773502691896258f;   // 3^-0.5

// Workspace layout (float offsets)
constexpr long OFF_A   = 0;                   // (B,NH,L,32) q-side [qs*SS | qp*SP*gamma | 0]
constexpr long OFF_B   = OFF_A   + 262144;    // (B,NH,L,32) k-side [ks | kp | 0]
constexpr long OFF_V   = OFF_B   + 262144;    // (B,NH,L,32) v-side [vs | vp | 0]
constexpr long OFF_QSQ = OFF_V   + 262144;    // (B,NH,L)  sum |q_p|^2
constexpr long OFF_KSQ = OFF_QSQ + 8192;      // (B,NH,L)  sum |k_p|^2
constexpr long OFF_ATT = OFF_KSQ + 8192;      // (B,NH,L,L) bias, then attn (in place)
constexpr long OFF_OPR = OFF_ATT + 4194304;   // (B,NH,L,12) raw attended points
constexpr long OFF_CAT = OFF_OPR + 98304;     // (B,L,1280) concatenated pre-output
// total: 6,406,144 floats = 25.6 MB

__device__ __forceinline__ v8f wmma_k4(float a0, float a1, float b0, float b1, v8f c) {
  v2f A; A[0] = a0; A[1] = a1;
  v2f B; B[0] = b0; B[1] = b1;
  // D = A(16x4,f32) * B(4x16,f32) + C  -> v_wmma_f32_16x16x4_f32
  return __builtin_amdgcn_wmma_f32_16x16x4_f32(false, A, false, B, (short)0, c, false, false);
}

// ---------------------------------------------------------------------------
// K1: projections + frame transforms + logit-operand packing
// grid = b*l blocks of 256
// ---------------------------------------------------------------------------
__global__ void k_proj(const float* __restrict__ x, const float* __restrict__ r,
                       const float* __restrict__ t,
                       const float* __restrict__ Wqs, const float* __restrict__ Wks,
                       const float* __restrict__ Wvs, const float* __restrict__ Wqp,
                       const float* __restrict__ Wkp, const float* __restrict__ Wvp,
                       const float* __restrict__ gamma, float* __restrict__ ws) {
  __shared__ float xs[DRES];
  __shared__ float proj[672];
  __shared__ float rr[9], tt[3];
  __shared__ float sqq[32], sqk[32];
  const int tid = threadIdx.x;
  const int bi  = blockIdx.x;        // b*512 + i
  const int b   = bi >> 9, i = bi & 511;
  for (int c = tid; c < DRES; c += 256) xs[c] = x[(long)bi * DRES + c];
  if (tid < 9) rr[tid] = r[bi * 9 + tid];
  if (tid >= 16 && tid < 19) tt[tid - 16] = t[bi * 3 + (tid - 16)];
  __syncthreads();
  for (int c = tid; c < 672; c += 256) {
    const float* W; int col, ncol;
    if      (c < 128) { W = Wqs; col = c;       ncol = 128; }
    else if (c < 256) { W = Wks; col = c - 128; ncol = 128; }
    else if (c < 384) { W = Wvs; col = c - 256; ncol = 128; }
    else if (c < 480) { W = Wqp; col = c - 384; ncol = 96;  }
    else if (c < 576) { W = Wkp; col = c - 480; ncol = 96;  }
    else              { W = Wvp; col = c - 576; ncol = 96;  }
    float acc = 0.f;
    for (int k = 0; k < DRES; ++k) acc += xs[k] * W[k * ncol + col];
    proj[c] = acc;
  }
  __syncthreads();
  float* Aq = ws + OFF_A; float* Bk = ws + OFF_B; float* Vv = ws + OFF_V;
  // scalar heads (dims 0..15)
  for (int c = tid; c < 128; c += 256) {
    const int n = c >> 4, d = c & 15;
    const long base = ((long)(b * NH + n) * LQ + i) * 32;
    Aq[base + d] = proj[c] * SS;
    Bk[base + d] = proj[128 + c];
    Vv[base + d] = proj[256 + c];
  }
  // points (dims 16..27): global = local @ r + t
  if (tid < 96) {
    const int role = tid / 32, np = tid & 31, n = np >> 2, p = np & 3;
    const float l0 = proj[384 + role * 96 + np * 3 + 0];
    const float l1 = proj[384 + role * 96 + np * 3 + 1];
    const float l2 = proj[384 + role * 96 + np * 3 + 2];
    float g[3];
#pragma unroll
    for (int c = 0; c < 3; ++c)
      g[c] = l0 * rr[0 * 3 + c] + l1 * rr[1 * 3 + c] + l2 * rr[2 * 3 + c] + tt[c];
    const long base = ((long)(b * NH + n) * LQ + i) * 32 + 16 + p * 3;
    const float s2 = g[0]*g[0] + g[1]*g[1] + g[2]*g[2];
    if (role == 0) {
      const float gs = SP * gamma[n];
      Aq[base + 0] = g[0] * gs; Aq[base + 1] = g[1] * gs; Aq[base + 2] = g[2] * gs;
      sqq[np] = s2;
    } else if (role == 1) {
      Bk[base + 0] = g[0]; Bk[base + 1] = g[1]; Bk[base + 2] = g[2];
      sqk[np] = s2;
    } else {
      Vv[base + 0] = g[0]; Vv[base + 1] = g[1]; Vv[base + 2] = g[2];
    }
  }
  // zero pad dims 28..31
  if (tid < 96) {
    const int arr = tid / 32, nd = tid & 31, n = nd >> 2, d = 28 + (nd & 3);
    const long base = ((long)(b * NH + n) * LQ + i) * 32 + d;
    (arr == 0 ? Aq : (arr == 1 ? Bk : Vv))[base] = 0.f;
  }
  __syncthreads();
  if (tid < NH) {
    ws[OFF_QSQ + (b * NH + tid) * LQ + i] =
        sqq[tid*4] + sqq[tid*4+1] + sqq[tid*4+2] + sqq[tid*4+3];
  } else if (tid < 2 * NH) {
    const int n = tid - NH;
    ws[OFF_KSQ + (b * NH + n) * LQ + i] =
        sqk[n*4] + sqk[n*4+1] + sqk[n*4+2] + sqk[n*4+3];
  }
}

// ---------------------------------------------------------------------------
// K2: bias(b,n,i,j) = e @ Wpb  via WMMA (M=16 pairs, N=16 (8 heads), K=128)
// grid = 2 * 1024 blocks of 256 (8 waves, each 2 M-tiles)
// ---------------------------------------------------------------------------
__global__ void k_bias(const float* __restrict__ e, const float* __restrict__ Wpb,
                       float* __restrict__ ws) {
  __shared__ float Wl[128][16];
  const int tid = threadIdx.x;
  for (int c = tid; c < 2048; c += 256) {
    const int k = c >> 4, n = c & 15;
    Wl[k][n] = (n < NH) ? Wpb[k * NH + n] : 0.f;
  }
  __syncthreads();
  const int b = blockIdx.x >> 10;
  const int pairBase = (blockIdx.x & 1023) * 256;
  const int w = tid >> 5, lane = tid & 31;
  const int half = lane >> 4, m16 = lane & 15;
  float* att = ws + OFF_ATT;
  for (int mt = 0; mt < 2; ++mt) {
    const int tbase = pairBase + w * 32 + mt * 16;
    const long erow = ((long)b * LQ * LQ + (tbase + m16)) * DPAIR;
    v8f acc = {};
    for (int k = 0; k < DPAIR; k += 4) {
      const int kk = k + 2 * half;
      const float2 av = *(const float2*)(e + erow + kk);
      acc = wmma_k4(av.x, av.y, Wl[kk][m16], Wl[kk + 1][m16], acc);
    }
    if (m16 < NH) {
#pragma unroll
      for (int v = 0; v < 8; ++v) {
        const int pr = tbase + v + 8 * half;
        const int i = pr >> 9, j = pr & 511;
        att[(((long)b * NH + m16) * LQ + i) * LQ + j] = acc[v];
      }
    }
  }
}

// ---------------------------------------------------------------------------
// K3: logits via WMMA (K=32: scalar+point operands) + bias + fused softmax
// grid = (b*NH, L/16) blocks of 256
// ---------------------------------------------------------------------------
__global__ void k_logits(const float* __restrict__ gamma, float* __restrict__ ws) {
  __shared__ float Al[16][34];
  __shared__ float lgts[16][512];
  __shared__ float qs16[16];
  __shared__ float red[16][16];
  __shared__ float rmax[16], rsum[16];
  const int tid = threadIdx.x;
  const int bn = blockIdx.x;            // b*8 + n
  const int n  = bn & 7;
  const int ibase = blockIdx.y * 16;
  const float* Aq  = ws + OFF_A;
  const float* Bk  = ws + OFF_B;
  const float* ksq = ws + OFF_KSQ;
  float* att = ws + OFF_ATT;
  for (int c = tid; c < 512; c += 256) {
    const int m = c >> 5, d = c & 31;
    Al[m][d] = Aq[((long)bn * LQ + ibase + m) * 32 + d];
  }
  if (tid < 16) qs16[tid] = ws[OFF_QSQ + bn * LQ + ibase + tid];
  __syncthreads();
  const float gcoef = -0.5f * SP * gamma[n];
  const int w = tid >> 5, lane = tid & 31;
  const int half = lane >> 4, m16 = lane & 15;
  for (int jt = w; jt < 32; jt += 8) {
    const int jb = jt * 16;
    v8f acc = {};
#pragma unroll
    for (int k = 0; k < 32; k += 4) {
      const int kk = k + 2 * half;
      const float2 av = *(const float2*)(&Al[m16][kk]);
      const float2 bv = *(const float2*)(Bk + ((long)bn * LQ + jb + m16) * 32 + kk);
      acc = wmma_k4(av.x, av.y, bv.x, bv.y, acc);
    }
    const int jj = jb + m16;
    const float kterm = ksq[bn * LQ + jj];
#pragma unroll
    for (int v = 0; v < 8; ++v) {
      const int m = v + 8 * half;
      const float bias = att[((long)bn * LQ + ibase + m) * LQ + jj];
      lgts[m][jj] = ST * (acc[v] + bias + gcoef * (qs16[m] + kterm));
    }
  }
  __syncthreads();
  const int r0 = tid >> 4, c0 = tid & 15;
  float mx = -3.4e38f;
  for (int j = c0; j < 512; j += 16) mx = fmaxf(mx, lgts[r0][j]);
  red[r0][c0] = mx;
  __syncthreads();
  if (c0 == 0) {
    float m2 = red[r0][0];
    for (int q = 1; q < 16; ++q) m2 = fmaxf(m2, red[r0][q]);
    rmax[r0] = m2;
  }
  __syncthreads();
  float s = 0.f;
  for (int j = c0; j < 512; j += 16) {
    const float ev = __expf(lgts[r0][j] - rmax[r0]);
    lgts[r0][j] = ev;
    s += ev;
  }
  red[r0][c0] = s;
  __syncthreads();
  if (c0 == 0) {
    float s2 = 0.f;
    for (int q = 0; q < 16; ++q) s2 += red[r0][q];
    rsum[r0] = s2;
  }
  __syncthreads();
  const float inv = 1.f / rsum[r0];
  for (int j = c0; j < 512; j += 16)
    att[((long)bn * LQ + ibase + r0) * LQ + j] = lgts[r0][j] * inv;
}

// ---------------------------------------------------------------------------
// K4: out_scalar & raw out_point = attn @ [v_s | v_p]  (M=16 i, N=32, K=512)
// grid = (b*NH, L/16) blocks of 64 (2 waves, one N-tile each)
// ---------------------------------------------------------------------------
__global__ void k_av(float* __restrict__ ws) {
  __shared__ float At[16][514];
  const int tid = threadIdx.x;   // 64 threads
  const int bn = blockIdx.x;
  const int b = bn >> 3, n = bn & 7;
  const int ibase = blockIdx.y * 16;
  const float* att = ws + OFF_ATT;
  const float* Vv  = ws + OFF_V;
  float* cat = ws + OFF_CAT;
  float* opr = ws + OFF_OPR;
  for (int c = tid; c < 16 * 512; c += 64) {
    const int m = c >> 9, j = c & 511;
    At[m][j] = att[((long)bn * LQ + ibase + m) * LQ + j];
  }
  __syncthreads();
  const int w = tid >> 5, lane = tid & 31;
  const int half = lane >> 4, m16 = lane & 15;
  const int col = w * 16 + m16;   // N column in [0,32)
  v8f acc = {};
  for (int k = 0; k < 512; k += 4) {
    const int kk = k + 2 * half;
    const float2 av = *(const float2*)(&At[m16][kk]);
    const float b0 = Vv[((long)bn * LQ + kk)     * 32 + col];
    const float b1 = Vv[((long)bn * LQ + kk + 1) * 32 + col];
    acc = wmma_k4(av.x, av.y, b0, b1, acc);
  }
#pragma unroll
  for (int v = 0; v < 8; ++v) {
    const int m = v + 8 * half;
    const int i = ibase + m;
    if (w == 0) {                                   // scalar head output
      cat[((long)b * LQ + i) * 1280 + n * 16 + m16] = acc[v];
    } else if (m16 < 12) {                          // attended points (raw)
      opr[((long)bn * LQ + i) * 12 + m16] = acc[v];
    }
  }
}

// ---------------------------------------------------------------------------
// K5: out_pair = attn @ e per (b,i)  (M=16 (8 heads), N=128, K=512)
// grid = b*l blocks of 256 (8 waves, one 16-wide d-tile each). e read once.
// ---------------------------------------------------------------------------
__global__ void k_ae(const float* __restrict__ e, float* __restrict__ ws) {
  __shared__ float Ar[16][514];
  const int tid = threadIdx.x;
  const int bi = blockIdx.x;
  const int b = bi >> 9, i = bi & 511;
  const float* att = ws + OFF_ATT;
  float* cat = ws + OFF_CAT;
  for (int c = tid; c < 16 * 512; c += 256) {
    const int m = c >> 9, j = c & 511;
    Ar[m][j] = (m < NH) ? att[(((long)b * NH + m) * LQ + i) * LQ + j] : 0.f;
  }
  __syncthreads();
  const int w = tid >> 5, lane = tid & 31;
  const int half = lane >> 4, m16 = lane & 15;
  const int dbase = w * 16;
  const long ebase = ((long)b * LQ + i) * LQ;
  v8f acc = {};
  for (int k = 0; k < 512; k += 4) {
    const int kk = k + 2 * half;
    const float2 av = *(const float2*)(&Ar[m16][kk]);
    const float b0 = e[(ebase + kk)     * DPAIR + dbase + m16];
    const float b1 = e[(ebase + kk + 1) * DPAIR + dbase + m16];
    acc = wmma_k4(av.x, av.y, b0, b1, acc);
  }
#pragma unroll
  for (int v = 0; v < 8; ++v) {
    const int m = v + 8 * half;
    if (m < NH)
      cat[((long)b * LQ + i) * 1280 + 128 + m * 128 + dbase + m16] = acc[v];
  }
}

// ---------------------------------------------------------------------------
// K6: inverse frame transform + point norms -> outcat
// ---------------------------------------------------------------------------
__global__ void k_points(const float* __restrict__ r, const float* __restrict__ t,
                         float* __restrict__ ws) {
  const int gid = blockIdx.x * 256 + threadIdx.x;  // [0, 2*512*8*4)
  if (gid >= 2 * LQ * NH * 4) return;
  const int p = gid & 3, n = (gid >> 2) & 7, i = (gid >> 5) & 511, b = gid >> 14;
  const float* opr = ws + OFF_OPR;
  float* cat = ws + OFF_CAT;
  const long obase = (((long)b * NH + n) * LQ + i) * 12 + p * 3;
  const int bi = b * LQ + i;
  float g[3], o[3];
#pragma unroll
  for (int c = 0; c < 3; ++c) g[c] = opr[obase + c] - t[bi * 3 + c];
#pragma unroll
  for (int c = 0; c < 3; ++c)
    o[c] = g[0] * r[bi * 9 + c * 3 + 0] + g[1] * r[bi * 9 + c * 3 + 1] +
           g[2] * r[bi * 9 + c * 3 + 2];
  const long cbase = ((long)b * LQ + i) * 1280;
  cat[cbase + 1152 + n * 12 + p * 3 + 0] = o[0];
  cat[cbase + 1152 + n * 12 + p * 3 + 1] = o[1];
  cat[cbase + 1152 + n * 12 + p * 3 + 2] = o[2];
  cat[cbase + 1248 + n * 4 + p] = sqrtf(o[0]*o[0] + o[1]*o[1] + o[2]*o[2]);
}

// ---------------------------------------------------------------------------
// K7: out = outcat @ Wout + bout  (M=1024, N=384, K=1280) via WMMA
// grid = (64, 3) blocks of 256 (8 waves, one 16-wide col-tile each)
// ---------------------------------------------------------------------------
__global__ void k_out(const float* __restrict__ Wout, const float* __restrict__ bout,
                      const float* __restrict__ ws, float* __restrict__ out) {
  const int tid = threadIdx.x;
  const int mt = blockIdx.x;       // row tile: rows [mt*16, mt*16+16)
  const int ng = blockIdx.y;       // col group of 128
  const int w = tid >> 5, lane = tid & 31;
  const int half = lane >> 4, m16 = lane & 15;
  const int col = ng * 128 + w * 16 + m16;
  const float* cat = ws + OFF_CAT;
  v8f acc = {};
  for (int k = 0; k < 1280; k += 4) {
    const int kk = k + 2 * half;
    const float2 av = *(const float2*)(cat + (long)(mt * 16 + m16) * 1280 + kk);
    const float b0 = Wout[(long)kk * DRES + col];
    const float b1 = Wout[(long)(kk + 1) * DRES + col];
    acc = wmma_k4(av.x, av.y, b0, b1, acc);
  }
  const float bb = bout[col];
#pragma unroll
  for (int v = 0; v < 8; ++v) {
    const int m = v + 8 * half;
    out[(long)(mt * 16 + m) * DRES + col] = acc[v] + bb;
  }
}

// ---------------------------------------------------------------------------
extern "C" void kernel_launch(void* const* d_in, const int* in_sizes, int n_in,
                              void* d_out, int out_size, void* d_ws, size_t ws_size,
                              hipStream_t stream) {
  const float* x     = (const float*)d_in[0];
  const float* e     = (const float*)d_in[1];
  const float* r     = (const float*)d_in[2];
  const float* t     = (const float*)d_in[3];
  const float* Wqs   = (const float*)d_in[4];
  const float* Wks   = (const float*)d_in[5];
  const float* Wvs   = (const float*)d_in[6];
  const float* Wpb   = (const float*)d_in[7];
  const float* Wqp   = (const float*)d_in[8];
  const float* Wkp   = (const float*)d_in[9];
  const float* Wvp   = (const float*)d_in[10];
  const float* gamma = (const float*)d_in[11];
  const float* Wout  = (const float*)d_in[12];
  const float* bout  = (const float*)d_in[13];
  float* ws  = (float*)d_ws;
  float* out = (float*)d_out;

  k_proj  <<<dim3(1024),   dim3(256), 0, stream>>>(x, r, t, Wqs, Wks, Wvs, Wqp, Wkp, Wvp, gamma, ws);
  k_bias  <<<dim3(2048),   dim3(256), 0, stream>>>(e, Wpb, ws);
  k_logits<<<dim3(16, 32), dim3(256), 0, stream>>>(gamma, ws);
  k_av    <<<dim3(16, 32), dim3(64),  0, stream>>>(ws);
  k_ae    <<<dim3(1024),   dim3(256), 0, stream>>>(e, ws);
  k_points<<<dim3(128),    dim3(256), 0, stream>>>(r, t, ws);
  k_out   <<<dim3(64, 3),  dim3(256), 0, stream>>>(Wout, bout, ws, out);
}